// ECHOBlock_6390911336487
// MI455X (gfx1250) — compile-verified
//
#include <hip/hip_runtime.h>
#include <math.h>

typedef float v2f __attribute__((ext_vector_type(2)));
typedef float v8f __attribute__((ext_vector_type(8)));

#define NN 20000
#define EE 640000
#define CC 8
#define DD 8
#define OO 64
#define RM 18          // R*M = 6*3
#define GWD 7
#define NDISK 37
#define MID 296        // D * NDISK
#define H1DIM 128
#define H2DIM 64
#define EPSF 1e-6f

// flat indices (i*7+j) of the disk (i-3)^2+(j-3)^2 <= 3.25^2 in a 7x7 grid
__constant__ int c_disk[NDISK] = {
    2,3,4,
    8,9,10,11,12,
    14,15,16,17,18,19,20,
    21,22,23,24,25,26,27,
    28,29,30,31,32,33,34,
    36,37,38,39,40,
    44,45,46};

// ---------------------------------------------------------------- utilities
__global__ void k_zero(float* p, long n) {
  long i = (long)blockIdx.x * blockDim.x + threadIdx.x;
  long stride = (long)gridDim.x * blockDim.x;
  for (; i < n; i += stride) p[i] = 0.f;
}

// Wc[d,c,rm] = W_conv[d,c,rm] * exp(i*phase[d,c])
__global__ void k_wc(const float* __restrict__ W_conv, const float* __restrict__ phase,
                     float* __restrict__ wc_re, float* __restrict__ wc_im) {
  int i = blockIdx.x * blockDim.x + threadIdx.x;
  if (i >= DD * CC * RM) return;
  int dc = i / RM;
  float ph = phase[dc];
  float w  = W_conv[i];
  wc_re[i] = w * cosf(ph);
  wc_im[i] = w * sinf(ph);
}

// G[n,d,rm] = sum_c Wc[d,c,rm] * x[n,c]   (complex) -- 23MB, L2 resident
__global__ void k_G(const float* __restrict__ xre, const float* __restrict__ xim,
                    const float* __restrict__ wcre, const float* __restrict__ wcim,
                    float* __restrict__ gre, float* __restrict__ gim) {
  int i = blockIdx.x * blockDim.x + threadIdx.x;
  if (i >= NN * DD * RM) return;
  int rm = i % RM;
  int nd = i / RM;
  int d  = nd % DD;
  int n  = nd / DD;
  float ar = 0.f, ai = 0.f;
#pragma unroll
  for (int c = 0; c < CC; ++c) {
    float wr = wcre[(d * CC + c) * RM + rm];
    float wi = wcim[(d * CC + c) * RM + rm];
    float xr = xre[n * CC + c];
    float xi = xim[n * CC + c];
    ar += wr * xr - wi * xi;
    ai += wr * xi + wi * xr;
  }
  gre[i] = ar;
  gim[i] = ai;
}

// ye[e,d] = sum_rm S[e,rm] * G[src_e,d,rm] ; scatter-add into y[dst]
__global__ void k_edge1(const int* __restrict__ edges,
                        const float* __restrict__ sre, const float* __restrict__ sim,
                        const float* __restrict__ gre, const float* __restrict__ gim,
                        float* __restrict__ yre, float* __restrict__ yim) {
  int e = blockIdx.x * blockDim.x + threadIdx.x;
  if (e >= EE) return;
  int src = edges[2 * e];
  int dst = edges[2 * e + 1];
  float s_re[RM], s_im[RM];
#pragma unroll
  for (int k = 0; k < RM; ++k) {
    s_re[k] = sre[(long)e * RM + k];
    s_im[k] = sim[(long)e * RM + k];
  }
  const float* gr = gre + (long)src * DD * RM;
  const float* gi = gim + (long)src * DD * RM;
#pragma unroll
  for (int d = 0; d < DD; ++d) {
    float ar = 0.f, ai = 0.f;
#pragma unroll
    for (int k = 0; k < RM; ++k) {
      float br = gr[d * RM + k];
      float bi = gi[d * RM + k];
      ar += s_re[k] * br - s_im[k] * bi;
      ai += s_re[k] * bi + s_im[k] * br;
    }
    atomicAdd(&yre[dst * DD + d], ar);
    atomicAdd(&yim[dst * DD + d], ai);
  }
}

// magnitude relu nonlinearity
__global__ void k_nl(const float* __restrict__ yre, const float* __restrict__ yim,
                     const float* __restrict__ b_nl,
                     float* __restrict__ zre, float* __restrict__ zim) {
  int i = blockIdx.x * blockDim.x + threadIdx.x;
  if (i >= NN * DD) return;
  int d = i % DD;
  float re = yre[i], im = yim[i];
  float mag = sqrtf(re * re + im * im);
  float s = fmaxf(mag + b_nl[d], 0.f) / (mag + EPSF);
  zre[i] = re * s;
  zim[i] = im * s;
}

// bilinear histogram scatter: hist[dst, gx, gy, d]
__global__ void k_hist(const int* __restrict__ edges,
                       const float* __restrict__ lnre, const float* __restrict__ lnim,
                       const float* __restrict__ wre, const float* __restrict__ wim,
                       const float* __restrict__ zre, const float* __restrict__ zim,
                       float* __restrict__ hist) {
  int e = blockIdx.x * blockDim.x + threadIdx.x;
  if (e >= EE) return;
  int src = edges[2 * e];
  int dst = edges[2 * e + 1];
  float lr = lnre[e], li = lnim[e];
  float pr = wre[e], pi = wim[e];
  float* hbase = hist + (long)dst * GWD * GWD * DD;
#pragma unroll
  for (int d = 0; d < DD; ++d) {
    float yr = zre[src * DD + d], yi = zim[src * DD + d];
    float fr = yr * pr - yi * pi;
    float fi = yr * pi + yi * pr;
    float fm = sqrtf(fr * fr + fi * fi);
    float inv = 1.f / (fm + EPSF);
    // coord = ln * conj(f) / (|f|+eps)
    float cr = (lr * fr + li * fi) * inv;
    float ci = (li * fr - lr * fi) * inv;
    float gx = cr * 3.f + 3.f;
    float gy = ci * 3.f + 3.f;
    float x0f = floorf(gx), y0f = floorf(gy);
    float wx = gx - x0f, wy = gy - y0f;
    int x0 = min(max((int)x0f, 0), GWD - 1);
    int x1 = min(x0 + 1, GWD - 1);
    int y0 = min(max((int)y0f, 0), GWD - 1);
    int y1 = min(y0 + 1, GWD - 1);
    atomicAdd(&hbase[(x0 * GWD + y0) * DD + d], fm * (1.f - wx) * (1.f - wy));
    atomicAdd(&hbase[(x1 * GWD + y0) * DD + d], fm * wx * (1.f - wy));
    atomicAdd(&hbase[(x0 * GWD + y1) * DD + d], fm * (1.f - wx) * wy);
    atomicAdd(&hbase[(x1 * GWD + y1) * DD + d], fm * wx * wy);
  }
}

// Fused 3-layer MLP + residual, 16 rows per workgroup, fp32 WMMA 16x16x4.
// A-layout (32-bit A 16x4): lane<16 -> K={0,1}, lane>=16 -> K={2,3}; M = lane&15.
// C/D layout: acc[v] is row (v + 8*(lane>=16)), col = lane&15 of the 16x16 tile.
__launch_bounds__(256)
__global__ void k_mlp(const float* __restrict__ hist,
                      const float* __restrict__ xre, const float* __restrict__ xim,
                      const float* __restrict__ W1, const float* __restrict__ b1,
                      const float* __restrict__ W2, const float* __restrict__ b2,
                      const float* __restrict__ W3, const float* __restrict__ b3,
                      const float* __restrict__ Wres, const float* __restrict__ bres,
                      float* __restrict__ out) {
  __shared__ float sdesc[16 * MID];
  __shared__ float sh1[16 * H1DIM];
  __shared__ float sh2[16 * H2DIM];
  __shared__ float sq[16 * CC];

  int tid  = threadIdx.x;
  int lane = tid & 31;
  int wave = tid >> 5;
  int n0   = blockIdx.x * 16;

  // gather desc tile through disk index: desc[n, d*37+s] = hist[n, disk[s], d]
  for (int t = tid; t < 16 * MID; t += 256) {
    int row = t / MID, k = t % MID;
    int d = k / NDISK, s = k % NDISK;
    sdesc[t] = hist[((long)(n0 + row) * (GWD * GWD) + c_disk[s]) * DD + d];
  }
  // residual feature q = sqrt(xre^2 + xim^2 + eps)
  for (int t = tid; t < 16 * CC; t += 256) {
    int row = t / CC, c = t % CC;
    float a = xre[(n0 + row) * CC + c];
    float b = xim[(n0 + row) * CC + c];
    sq[t] = sqrtf(a * a + b * b + EPSF);
  }
  __syncthreads();

  int mrow  = lane & 15;
  int khalf = (lane >> 4) * 2;  // 0 or 2
  int vrow0 = (lane >> 4) << 3; // 0 or 8

  // ---- layer 1: 296 -> 128, each of 8 waves owns 16 output cols
  {
    int col = wave * 16 + (lane & 15);
    v8f acc = {};
    for (int kc = 0; kc < MID / 4; ++kc) {   // 296 = 74*4 exactly
      int k = kc * 4 + khalf;
      v2f a, b;
      a.x = sdesc[mrow * MID + k];
      a.y = sdesc[mrow * MID + k + 1];
      b.x = W1[col * MID + k];
      b.y = W1[col * MID + k + 1];
      acc = __builtin_amdgcn_wmma_f32_16x16x4_f32(false, a, false, b,
                                                  (short)0, acc, false, false);
    }
    float bias = b1[col];
#pragma unroll
    for (int v = 0; v < 8; ++v)
      sh1[(v + vrow0) * H1DIM + col] = fmaxf(acc[v] + bias, 0.f);
  }
  __syncthreads();

  // ---- layer 2: 128 -> 64, waves 0..3
  if (wave < 4) {
    int col = wave * 16 + (lane & 15);
    v8f acc = {};
    for (int kc = 0; kc < H1DIM / 4; ++kc) {
      int k = kc * 4 + khalf;
      v2f a, b;
      a.x = sh1[mrow * H1DIM + k];
      a.y = sh1[mrow * H1DIM + k + 1];
      b.x = W2[col * H1DIM + k];
      b.y = W2[col * H1DIM + k + 1];
      acc = __builtin_amdgcn_wmma_f32_16x16x4_f32(false, a, false, b,
                                                  (short)0, acc, false, false);
    }
    float bias = b2[col];
#pragma unroll
    for (int v = 0; v < 8; ++v)
      sh2[(v + vrow0) * H2DIM + col] = fmaxf(acc[v] + bias, 0.f);
  }
  __syncthreads();

  // ---- layer 3: 64 -> 64 (+ residual), waves 0..3
  if (wave < 4) {
    int col = wave * 16 + (lane & 15);
    v8f acc = {};
    for (int kc = 0; kc < H2DIM / 4; ++kc) {
      int k = kc * 4 + khalf;
      v2f a, b;
      a.x = sh2[mrow * H2DIM + k];
      a.y = sh2[mrow * H2DIM + k + 1];
      b.x = W3[col * H2DIM + k];
      b.y = W3[col * H2DIM + k + 1];
      acc = __builtin_amdgcn_wmma_f32_16x16x4_f32(false, a, false, b,
                                                  (short)0, acc, false, false);
    }
    float bias = b3[col] + bres[col];
#pragma unroll
    for (int v = 0; v < 8; ++v) {
      int row = v + vrow0;
      float r = 0.f;
#pragma unroll
      for (int c = 0; c < CC; ++c) r += sq[row * CC + c] * Wres[col * CC + c];
      out[(long)(n0 + row) * OO + col] = acc[v] + bias + r;
    }
  }
}

extern "C" void kernel_launch(void* const* d_in, const int* in_sizes, int n_in,
                              void* d_out, int out_size, void* d_ws, size_t ws_size,
                              hipStream_t stream) {
  const float* x_re    = (const float*)d_in[0];
  const float* x_im    = (const float*)d_in[1];
  const int*   edges   = (const int*)  d_in[2];
  const float* sten_re = (const float*)d_in[3];
  const float* sten_im = (const float*)d_in[4];
  const float* ln_re   = (const float*)d_in[5];
  const float* ln_im   = (const float*)d_in[6];
  const float* wxp_re  = (const float*)d_in[7];
  const float* wxp_im  = (const float*)d_in[8];
  const float* W_conv  = (const float*)d_in[9];
  const float* phase   = (const float*)d_in[10];
  const float* b_nl    = (const float*)d_in[11];
  const float* W1      = (const float*)d_in[12];
  const float* b1      = (const float*)d_in[13];
  const float* W2      = (const float*)d_in[14];
  const float* b2      = (const float*)d_in[15];
  const float* W3      = (const float*)d_in[16];
  const float* b3      = (const float*)d_in[17];
  const float* Wres    = (const float*)d_in[18];
  const float* bres    = (const float*)d_in[19];
  float* out = (float*)d_out;

  float* ws = (float*)d_ws;
  size_t off = 0;
  float* wc_re = ws + off; off += DD * CC * RM;
  float* wc_im = ws + off; off += DD * CC * RM;
  float* g_re  = ws + off; off += (size_t)NN * DD * RM;
  float* g_im  = ws + off; off += (size_t)NN * DD * RM;
  float* y_re  = ws + off; off += NN * DD;
  float* y_im  = ws + off; off += NN * DD;
  float* z_re  = ws + off; off += NN * DD;
  float* z_im  = ws + off; off += NN * DD;
  float* hist  = ws + off; off += (size_t)NN * GWD * GWD * DD;

  // zero accumulator range (y_re .. end of hist) every call
  long nz = (long)(4 * NN * DD) + (long)NN * GWD * GWD * DD;
  k_zero<<<4096, 256, 0, stream>>>(y_re, nz);

  k_wc<<<(DD * CC * RM + 255) / 256, 256, 0, stream>>>(W_conv, phase, wc_re, wc_im);
  k_G<<<(NN * DD * RM + 255) / 256, 256, 0, stream>>>(x_re, x_im, wc_re, wc_im, g_re, g_im);
  k_edge1<<<(EE + 255) / 256, 256, 0, stream>>>(edges, sten_re, sten_im, g_re, g_im, y_re, y_im);
  k_nl<<<(NN * DD + 255) / 256, 256, 0, stream>>>(y_re, y_im, b_nl, z_re, z_im);
  k_hist<<<(EE + 255) / 256, 256, 0, stream>>>(edges, ln_re, ln_im, wxp_re, wxp_im,
                                               z_re, z_im, hist);
  k_mlp<<<NN / 16, 256, 0, stream>>>(hist, x_re, x_im, W1, b1, W2, b2, W3, b3,
                                     Wres, bres, out);
}